// LinearChainCRF_214748365396
// MI455X (gfx1250) — compile-verified
//
#include <hip/hip_runtime.h>
#include <hip/hip_bf16.h>
#include <cstdint>
#include <cstddef>

// Linear-chain CRF forward (scaled exp-space recursion) for MI455X / gfx1250.
// Each step: U = (Q * E) ∘ exp(em_t + start), E = exp(trans) (0 where forbidden).
// Q (128 rows x 128 states) is tiled 16x16 per wave; E B-fragments are
// loop-invariant and stay in VGPRs; the 128x128x128 f32 matmul per step is
// done with v_wmma_f32_16x16x4_f32.

typedef __attribute__((ext_vector_type(2))) float v2f;
typedef __attribute__((ext_vector_type(8))) float v8f;

static constexpr int Bn   = 64;
static constexpr int Tn   = 2048;
static constexpr int Kn   = 128;
static constexpr int PSTR = 132;   // padded row stride: 132 % 64 banks -> 4r pattern, conflict-free b64 loads

__global__ __launch_bounds__(256) void crf_forward_kernel(
    const float*   __restrict__ em,     // (B,T,K) f32
    const uint8_t* __restrict__ mask,   // (B,T)   bool
    const uint8_t* __restrict__ tgt,    // (B,T,K) bool
    const float*   __restrict__ trans,  // (K,K)   f32
    const float*   __restrict__ stt,    // (K,)    f32
    const uint8_t* __restrict__ forb,   // (K,K)   bool
    float*         __restrict__ zbuf)   // (128,)  per-row log-partition
{
    __shared__ float p_lds[2][16 * PSTR];   // double-buffered Q tiles (16 rows x 128 states)
    __shared__ float partial[16][8];        // per-wave partial row sums (deterministic tree)
    __shared__ float rowsum[16];            // S_{t-1} per row
    __shared__ float c_lds[16];             // running log-normalizer per row
    __shared__ int   len_lds[16];           // sequence lengths

    const int tid   = threadIdx.x;
    const int lane  = tid & 31;
    const int wave  = tid >> 5;          // 0..7 -> column tile
    const int lm    = lane & 15;
    const int hs    = lane >> 4;         // half-select (lanes 16-31)
    const int colN  = wave * 16 + lm;    // this lane's N column (0..127)
    const int Rbase = blockIdx.x * 16;   // global row base; rows < 64 are supervised
    const bool sup  = (Rbase < Bn);

    // ---- one-time init: lengths (popcount of mask bytes), c, rowsum ----
    if (tid < 16) {
        c_lds[tid]  = 0.0f;
        rowsum[tid] = 0.0f;
        const int b = (Rbase + tid) & (Bn - 1);
        const unsigned long long* m8 =
            (const unsigned long long*)(mask + (size_t)b * Tn);
        int l = 0;
        for (int i = 0; i < Tn / 8; ++i) l += __popcll(m8[i]);
        len_lds[tid] = l;
    }

    // ---- preload loop-invariant B-fragments of E = exp(trans), 0 if forbidden ----
    v2f Bf[32];
#pragma unroll
    for (int kc = 0; kc < 32; ++kc) {
        const int k0 = 4 * kc + 2 * hs;
        const size_t i0 = (size_t)k0 * Kn + colN;
        const size_t i1 = i0 + Kn;
        Bf[kc].x = forb[i0] ? 0.0f : expf(trans[i0]);
        Bf[kc].y = forb[i1] ? 0.0f : expf(trans[i1]);
    }
    const float st = stt[colN];

    // emission/target base pointers for this lane's 8 (row M = v + 8*hs, colN) slots
    const float*   emp[8];
    const uint8_t* tgp[8];
#pragma unroll
    for (int v = 0; v < 8; ++v) {
        const int R = Rbase + v + 8 * hs;
        const int b = R & (Bn - 1);
        emp[v] = em  + (size_t)b * Tn * Kn + colN;
        tgp[v] = tgt + (size_t)b * Tn * Kn + colN;
    }

    // prologue loads for t = 0
    float emc[8];
    int   tgc[8];
#pragma unroll
    for (int v = 0; v < 8; ++v) {
        emc[v] = emp[v][0];
        tgc[v] = sup ? (int)tgp[v][0] : 1;
    }

    __syncthreads();

    for (int t = 0; t < Tn; ++t) {
        const int wb = t & 1;        // write buffer
        const int rb = wb ^ 1;       // read buffer (holds Q(t-1))

        // -- software prefetch emissions/targets for t+1 (clamped) --
        float emn[8];
        int   tgn[8];
        {
            const int tn = (t + 1 < Tn) ? (t + 1) : (Tn - 1);
            const size_t off = (size_t)tn * Kn;
#pragma unroll
            for (int v = 0; v < 8; ++v) {
                emn[v] = emp[v][off];
                tgn[v] = sup ? (int)tgp[v][off] : 1;
            }
        }

        // -- phase A: U = (Q·E) * rcp(S_prev) * exp(em + start) --
        float u[8];
        if (t > 0) {
            v8f a0c = {}, a1c = {}, a2c = {}, a3c = {};
#pragma unroll
            for (int kc = 0; kc < 32; kc += 4) {
                const float* base = &p_lds[rb][lm * PSTR + 2 * hs];
                v2f a0 = *(const v2f*)(base + 4 * (kc + 0));
                v2f a1 = *(const v2f*)(base + 4 * (kc + 1));
                v2f a2 = *(const v2f*)(base + 4 * (kc + 2));
                v2f a3 = *(const v2f*)(base + 4 * (kc + 3));
                a0c = __builtin_amdgcn_wmma_f32_16x16x4_f32(false, a0, false, Bf[kc + 0], (short)0, a0c, false, false);
                a1c = __builtin_amdgcn_wmma_f32_16x16x4_f32(false, a1, false, Bf[kc + 1], (short)0, a1c, false, false);
                a2c = __builtin_amdgcn_wmma_f32_16x16x4_f32(false, a2, false, Bf[kc + 2], (short)0, a2c, false, false);
                a3c = __builtin_amdgcn_wmma_f32_16x16x4_f32(false, a3, false, Bf[kc + 3], (short)0, a3c, false, false);
            }
            const v8f D = (a0c + a1c) + (a2c + a3c);
#pragma unroll
            for (int v = 0; v < 8; ++v) {
                const float S = rowsum[v + 8 * hs];        // S_{t-1} for this row
                float f = __builtin_amdgcn_rcpf(S);
                f = f * (2.0f - S * f);                    // Newton refine
                const float e = tgc[v] ? expf(emc[v] + st) : 0.0f;
                u[v] = D[v] * f * e;
            }
        } else {
#pragma unroll
            for (int v = 0; v < 8; ++v)
                u[v] = tgc[v] ? expf(emc[v] + st) : 0.0f;
        }

        // store U tile into write buffer (becomes next step's Q)
#pragma unroll
        for (int v = 0; v < 8; ++v)
            p_lds[wb][(v + 8 * hs) * PSTR + colN] = u[v];

        // per-wave deterministic row sums over this wave's 16 columns
#pragma unroll
        for (int v = 0; v < 8; ++v) {
            float val = u[v];
            val += __shfl_xor(val, 1, 32);
            val += __shfl_xor(val, 2, 32);
            val += __shfl_xor(val, 4, 32);
            val += __shfl_xor(val, 8, 32);
            if (lm == 0) partial[v + 8 * hs][wave] = val;   // lanes 0 and 16
        }

        __syncthreads();

        // -- phase B: fixed-order total sums, normalizer update, record z --
        if (tid < 16) {
            float S = 0.0f;
#pragma unroll
            for (int w = 0; w < 8; ++w) S += partial[tid][w];
            rowsum[tid] = S;
            const float cc = c_lds[tid] + logf(S);
            c_lds[tid] = cc;
            if (t == len_lds[tid] - 1) zbuf[Rbase + tid] = cc;
        }

        __syncthreads();

#pragma unroll
        for (int v = 0; v < 8; ++v) { emc[v] = emn[v]; tgc[v] = tgn[v]; }
    }
}

__global__ void crf_diff_kernel(const float* __restrict__ zbuf,
                                float* __restrict__ out)
{
    const int b = threadIdx.x;
    if (b < Bn) out[b] = zbuf[b] - zbuf[b + Bn];
}

extern "C" void kernel_launch(void* const* d_in, const int* in_sizes, int n_in,
                              void* d_out, int out_size, void* d_ws, size_t ws_size,
                              hipStream_t stream)
{
    (void)in_sizes; (void)n_in; (void)out_size; (void)ws_size;
    const float*   em    = (const float*)  d_in[0];   // emissions (B,T,K) f32
    const uint8_t* mask  = (const uint8_t*)d_in[1];   // mask (B,T) bool
    const uint8_t* tgt   = (const uint8_t*)d_in[2];   // target (B,T,K) bool
    const float*   trans = (const float*)  d_in[3];   // transitions (K,K) f32
    const float*   stt   = (const float*)  d_in[4];   // start_transitions (K,) f32
    const uint8_t* forb  = (const uint8_t*)d_in[5];   // forbidden (K,K) bool

    float* zbuf = (float*)d_ws;                       // 128 floats of scratch

    crf_forward_kernel<<<8, 256, 0, stream>>>(em, mask, tgt, trans, stt, forb, zbuf);
    crf_diff_kernel<<<1, 64, 0, stream>>>(zbuf, (float*)d_out);
}